// TextRNN_68719477193
// MI455X (gfx1250) — compile-verified
//
#include <hip/hip_runtime.h>
#include <math.h>

// ---------------------------------------------------------------------------
// TextRNN on MI455X (gfx1250): 256-step tanh RNN via fp32 WMMA (16x16x4),
// double-buffered LDS staging with async global->LDS copies (ASYNCcnt).
// ---------------------------------------------------------------------------

typedef __attribute__((ext_vector_type(2))) float v2f;
typedef __attribute__((ext_vector_type(8))) float v8f;
typedef int v4i_t __attribute__((vector_size(16)));   // builtin's pointee type

#define BATCH   256
#define SEQ     256
#define EMBED   512
#define HIDDEN  1024
#define NOUT    20

// LDS row stride in floats: even (8B-aligned v2f frag reads), multiple of 4
// (16B-aligned b128 stores), and 36r mod 64 distinct for r=0..15 -> no bank
// conflicts on fragment reads.
#define LDS_STRIDE 36

#if defined(__HIP_DEVICE_COMPILE__) && \
    __has_builtin(__builtin_amdgcn_global_load_async_to_lds_b128) && \
    __has_builtin(__builtin_amdgcn_s_wait_asynccnt)
#define USE_ASYNC_LDS 1
#else
#define USE_ASYNC_LDS 0
#endif

#if USE_ASYNC_LDS
// 16B global -> LDS, no VGPR round trip; tracked by ASYNCcnt.
#define COPY16(gp, lp)                                                        \
  __builtin_amdgcn_global_load_async_to_lds_b128(                             \
      (v4i_t*)(void*)(gp), (v4i_t*)(void*)(lp), 0, 0)
#define ASYNC_WAIT(n) __builtin_amdgcn_s_wait_asynccnt(n)
#else
#define COPY16(gp, lp) (*(float4*)(lp) = *(const float4*)(gp))
#define ASYNC_WAIT(n)  ((void)0)
#endif

// One recurrence step:  h_next = tanh( gather(emb,inputs[:,s]) @ Wx^T
//                                      + (first ? 0 : h_prev @ Wh^T)
//                                      + bx + bh )
// Grid: (HIDDEN/128, BATCH/64), block 256 threads = 8 wave32s (2 M x 4 N),
// wave tile 32x32 = 2x2 fragments of 16x16.  Unified K pipeline:
// chunks 0..15  : K-slice of the gathered-embedding @ Wx^T product
// chunks 16..47 : K-slice of h_prev @ Wh^T (skipped when first==1)
__global__ __launch_bounds__(256) void rnn_step_kernel(
    const int*   __restrict__ inputs,   // [BATCH, SEQ]
    const float* __restrict__ emb,      // [VOCAB, EMBED]
    const float* __restrict__ Wx,       // [HIDDEN, EMBED]
    const float* __restrict__ bx,       // [HIDDEN]
    const float* __restrict__ Wh,       // [HIDDEN, HIDDEN]
    const float* __restrict__ bh,       // [HIDDEN]
    const float* __restrict__ h_prev,   // [BATCH, HIDDEN]
    float*       __restrict__ h_next,   // [BATCH, HIDDEN]
    int s, int first)
{
  __shared__ float As[2 * 64  * LDS_STRIDE];
  __shared__ float Bs[2 * 128 * LDS_STRIDE];
  __shared__ int   sIdx[64];

  const int tid  = threadIdx.x;
  const int lane = tid & 31;          // wave32
  const int wave = tid >> 5;          // 0..7
  const int wm   = wave >> 2;         // 0..1  (M direction)
  const int wn   = wave & 3;          // 0..3  (N direction)
  const int l15  = lane & 15;
  const int lhi  = lane >> 4;         // 0/1

  const int bbase = blockIdx.y * 64;   // batch-row base
  const int nbase = blockIdx.x * 128;  // hidden-col base

  if (tid < 64) sIdx[tid] = inputs[(size_t)(bbase + tid) * SEQ + s];
  __syncthreads();   // sIdx visible before chunk-0 gather

  v8f acc[2][2] = {};

  const int nk = first ? 16 : 48;      // 16 x-chunks (+ 32 h-chunks)

  // Issue one 32-wide K-chunk of A(64 rows) and B(128 rows) into LDS buffer.
  // Per wave: 2 A ops + 4 B ops = 6 async b128 instructions.
  auto issue_chunk = [&](int c, int buf) {
    float* Asb = As + buf * (64  * LDS_STRIDE);
    float* Bsb = Bs + buf * (128 * LDS_STRIDE);
    const int ph2 = (c >= 16);
    const int kb  = (ph2 ? (c - 16) : c) * 32;
    #pragma unroll
    for (int i = tid; i < 64 * 8; i += 256) {       // A: 64 x 32 floats
      const int r = i >> 3, cc = (i & 7) << 2;
      const float* gp = ph2
          ? h_prev + (size_t)(bbase + r) * HIDDEN + kb + cc
          : emb    + (size_t)sIdx[r]     * EMBED  + kb + cc;
      COPY16(gp, Asb + r * LDS_STRIDE + cc);
    }
    #pragma unroll
    for (int i = tid; i < 128 * 8; i += 256) {      // B: 128 x 32 floats
      const int r = i >> 3, cc = (i & 7) << 2;
      const float* gp = ph2
          ? Wh + (size_t)(nbase + r) * HIDDEN + kb + cc
          : Wx + (size_t)(nbase + r) * EMBED  + kb + cc;
      COPY16(gp, Bsb + r * LDS_STRIDE + cc);
    }
  };

  // 8 K-steps of 4 per chunk; f32 WMMA fragments:
  //   A 16x4: lane -> (row = lane&15, k = kk + 2*(lane>>4)) float2
  //   B 4x16: lane -> (col = lane&15, k = kk + 2*(lane>>4)) float2
  auto mma_chunk = [&](int buf) {
    const float* Asb = As + buf * (64  * LDS_STRIDE);
    const float* Bsb = Bs + buf * (128 * LDS_STRIDE);
    #pragma unroll
    for (int kk = 0; kk < 32; kk += 4) {
      v2f a[2], b[2];
      #pragma unroll
      for (int mt = 0; mt < 2; ++mt)
        a[mt] = *(const v2f*)(Asb + (wm * 32 + mt * 16 + l15) * LDS_STRIDE + kk + 2 * lhi);
      #pragma unroll
      for (int nt = 0; nt < 2; ++nt)
        b[nt] = *(const v2f*)(Bsb + (wn * 32 + nt * 16 + l15) * LDS_STRIDE + kk + 2 * lhi);
      #pragma unroll
      for (int mt = 0; mt < 2; ++mt)
        #pragma unroll
        for (int nt = 0; nt < 2; ++nt)
          acc[mt][nt] = __builtin_amdgcn_wmma_f32_16x16x4_f32(
              false, a[mt], false, b[nt], (short)0, acc[mt][nt], false, false);
    }
  };

  // Software-pipelined double buffer:
  //   issue(c+1) -> wait(older done, 6 newest may stay outstanding) ->
  //   barrier -> mma(c) -> barrier (buf c reusable for chunk c+2)
  issue_chunk(0, 0);
  for (int c = 0; c < nk; ++c) {
    const int cur = c & 1;
    if (c + 1 < nk) {
      issue_chunk(c + 1, cur ^ 1);
      ASYNC_WAIT(6);            // chunk c complete (in-order per wave)
    } else {
      ASYNC_WAIT(0);
    }
    __syncthreads();            // all waves' chunk-c data visible in LDS
    mma_chunk(cur);
    __syncthreads();            // all reads of buf `cur` done before rewrite
  }

  // ---------------- epilogue: tanh(acc + bx + bh) --------------------------
  // C/D layout: VGPR g -> M = g + 8*(lane>>4), N = lane&15.
  #pragma unroll
  for (int mt = 0; mt < 2; ++mt) {
    #pragma unroll
    for (int nt = 0; nt < 2; ++nt) {
      const int col  = nbase + wn * 32 + nt * 16 + l15;
      const float bsum = bx[col] + bh[col];
      #pragma unroll
      for (int g = 0; g < 8; ++g) {
        const int row = bbase + wm * 32 + mt * 16 + g + (lhi << 3);
        h_next[(size_t)row * HIDDEN + col] = tanhf(acc[mt][nt][g] + bsum);
      }
    }
  }
}

// Final classifier: logits = h @ Wy^T + by ; out = log_softmax(logits).
// One block (128 threads) per batch row; NOUT=20 is tiny.
__global__ __launch_bounds__(128) void logits_kernel(
    const float* __restrict__ h,    // [BATCH, HIDDEN]
    const float* __restrict__ Wy,   // [NOUT, HIDDEN]
    const float* __restrict__ by,   // [NOUT]
    float*       __restrict__ out)  // [BATCH, NOUT]
{
  const int b = blockIdx.x;
  __shared__ float red[NOUT];
  if (threadIdx.x < NOUT) red[threadIdx.x] = 0.0f;
  __syncthreads();

  float partial[NOUT];
  #pragma unroll
  for (int o = 0; o < NOUT; ++o) partial[o] = 0.0f;

  for (int k = threadIdx.x; k < HIDDEN; k += 128) {
    const float hv = h[(size_t)b * HIDDEN + k];
    #pragma unroll
    for (int o = 0; o < NOUT; ++o)
      partial[o] += hv * Wy[(size_t)o * HIDDEN + k];
  }
  #pragma unroll
  for (int o = 0; o < NOUT; ++o) atomicAdd(&red[o], partial[o]);
  __syncthreads();

  if (threadIdx.x == 0) {
    float t[NOUT], m = -1e30f;
    #pragma unroll
    for (int o = 0; o < NOUT; ++o) { t[o] = red[o] + by[o]; m = fmaxf(m, t[o]); }
    float ssum = 0.0f;
    #pragma unroll
    for (int o = 0; o < NOUT; ++o) ssum += __expf(t[o] - m);
    const float lse = m + logf(ssum);
    #pragma unroll
    for (int o = 0; o < NOUT; ++o) out[(size_t)b * NOUT + o] = t[o] - lse;
  }
}

extern "C" void kernel_launch(void* const* d_in, const int* in_sizes, int n_in,
                              void* d_out, int out_size, void* d_ws, size_t ws_size,
                              hipStream_t stream) {
  const int*   inputs = (const int*)  d_in[0];
  const float* emb    = (const float*)d_in[1];
  const float* Wx     = (const float*)d_in[2];
  const float* bxv    = (const float*)d_in[3];
  const float* Wh     = (const float*)d_in[4];
  const float* bhv    = (const float*)d_in[5];
  const float* Wy     = (const float*)d_in[6];
  const float* byv    = (const float*)d_in[7];
  float* out = (float*)d_out;

  float* hA = (float*)d_ws;
  float* hB = hA + (size_t)BATCH * HIDDEN;
  float* prev = hA;
  float* next = hB;

  const dim3 grid(HIDDEN / 128, BATCH / 64);  // (8, 4)
  for (int s = 0; s < SEQ; ++s) {
    rnn_step_kernel<<<grid, 256, 0, stream>>>(inputs, emb, Wx, bxv, Wh, bhv,
                                              prev, next, s, (s == 0) ? 1 : 0);
    float* t = prev; prev = next; next = t;
  }
  logits_kernel<<<dim3(BATCH), 128, 0, stream>>>(prev, Wy, byv, out);
}